// GATEncoder_44659069944069
// MI455X (gfx1250) — compile-verified
//
#include <hip/hip_runtime.h>

typedef __attribute__((ext_vector_type(2))) float v2f;
typedef __attribute__((ext_vector_type(8))) float v8f;

#define NEG_SLOPE 0.2f
#define EPS_F 1e-16f

// ---- monotone float<->uint mapping for atomic segment-max -------------------
__device__ __forceinline__ unsigned fenc(float f) {
  unsigned u = __float_as_uint(f);
  return (u & 0x80000000u) ? ~u : (u | 0x80000000u);
}
__device__ __forceinline__ float fdec(unsigned u) {
  return (u & 0x80000000u) ? __uint_as_float(u & 0x7fffffffu)
                           : __uint_as_float(~u);
}

// ---- WMMA GEMM: XS[N,128] = X[N,K] * W[128,K]^T  (fp32, 16x16x4) ------------
// W is staged into LDS once per block (padded pitch K+4 -> bank-conflict-free
// column reads), then each wave computes one 16-row tile: 8 N-tiles x (K/4)
// v_wmma_f32_16x16x4_f32 steps. ISA 7.12.2 fp32 layouts:
//   A 16x4 : lane holds A[m][2*kh + {0,1}], m=lane&15, kh=lane>>4
//   B 4x16 : lane holds B[2*kh + {0,1}][n], n=lane&15
//   C 16x16: VGPR r -> row (r + 8*kh), col (lane&15)
template <int K>
__global__ __launch_bounds__(256) void gemm_node_wmma(
    const float* __restrict__ X, const float* __restrict__ W,
    float* __restrict__ XS, int Nrows) {
  constexpr int KP = K + 4;                 // padded LDS pitch (floats)
  __shared__ float wlds[128 * KP];

  // ---- cooperative stage of W[128,K] into LDS (float4 rows) ----
  {
    constexpr int R4 = K / 4;               // float4s per row
    const float4* wsrc = (const float4*)W;
    for (int i = threadIdx.x; i < 128 * R4; i += 256) {
      int n = i / R4, k4 = i % R4;
      *(float4*)(wlds + n * KP + k4 * 4) = wsrc[i];
    }
  }
  __syncthreads();

  const int lane = threadIdx.x & 31;
  const int wave = threadIdx.x >> 5;
  const int mtile = blockIdx.x * 8 + wave;
  const int m0 = mtile * 16;
  if (m0 >= Nrows) return;

  const int mr = lane & 15;
  const int kh = lane >> 4;  // 0 or 1
  int row = m0 + mr;
  if (row >= Nrows) row = Nrows - 1;        // clamp (EXEC stays all-ones)

  constexpr int KS = K / 4;
  float a0[KS], a1[KS];
  {
    const float* xp = X + (size_t)row * K + 2 * kh;
#pragma unroll
    for (int kk = 0; kk < KS; ++kk) {
      a0[kk] = xp[kk * 4];
      a1[kk] = xp[kk * 4 + 1];
    }
  }

  const float* lptr = wlds + mr * KP + 2 * kh;   // per-lane B base in LDS
  float* outp = XS + (size_t)(m0 + 8 * kh) * 128 + mr;
  const bool full = (m0 + 16 <= Nrows);

#pragma unroll
  for (int nt = 0; nt < 8; ++nt) {
    v8f acc = {0.f, 0.f, 0.f, 0.f, 0.f, 0.f, 0.f, 0.f};
#pragma unroll
    for (int kk = 0; kk < KS; ++kk) {
      v2f A;
      A[0] = a0[kk];
      A[1] = a1[kk];
      const float* wp = lptr + nt * 16 * KP + kk * 4;  // immediate DS offsets
      v2f Bv;
      Bv[0] = wp[0];
      Bv[1] = wp[1];
      acc = __builtin_amdgcn_wmma_f32_16x16x4_f32(
          /*neg_a=*/false, A, /*neg_b=*/false, Bv,
          /*c_mod=*/(short)0, acc, /*reuse_a=*/false, /*reuse_b=*/false);
    }
    if (full) {
#pragma unroll
      for (int r = 0; r < 8; ++r) outp[r * 128 + nt * 16] = acc[r];
    } else {
#pragma unroll
      for (int r = 0; r < 8; ++r)
        if (m0 + r + 8 * kh < Nrows) outp[r * 128 + nt * 16] = acc[r];
    }
  }
}

// ---- self-loop mean attr ----------------------------------------------------
__global__ __launch_bounds__(256) void k_edge_degree(
    const int* __restrict__ dst, const float* __restrict__ ew,
    float* __restrict__ sums, float* __restrict__ cnt, int E) {
  int i = blockIdx.x * blockDim.x + threadIdx.x;
  if (i < E) {
    int d = dst[i];
    atomicAdd(&sums[d], ew[i]);
    atomicAdd(&cnt[d], 1.0f);
  }
}

__global__ __launch_bounds__(256) void k_build_ea(
    const float* __restrict__ ew, const float* __restrict__ sums,
    const float* __restrict__ cnt, float* __restrict__ ea, int E, int EN) {
  int i = blockIdx.x * blockDim.x + threadIdx.x;
  if (i < EN)
    ea[i] = (i < E) ? ew[i] : sums[i - E] / fmaxf(cnt[i - E], 1.0f);
}

// wdot[h] = sum_c we[h*32+c]*ae[h*32+c] for both layers
__global__ void k_wedot(const float* __restrict__ we1,
                        const float* __restrict__ ae1,
                        const float* __restrict__ we2,
                        const float* __restrict__ ae2,
                        float* __restrict__ wdot) {
  int h = threadIdx.x;
  if (h < 4) {
    float s1 = 0.f, s2 = 0.f;
    for (int c = 0; c < 32; ++c) {
      s1 += we1[h * 32 + c] * ae1[h * 32 + c];
      s2 += we2[h * 32 + c] * ae2[h * 32 + c];
    }
    wdot[h] = s1;
    wdot[4 + h] = s2;
  }
}

// a_src[n,h], a_dst[n,h] from xs[n, h*32+c]
__global__ __launch_bounds__(256) void k_node_attn(
    const float* __restrict__ xs, const float* __restrict__ as_,
    const float* __restrict__ ad_, float* __restrict__ asrc,
    float* __restrict__ adst, int N) {
  int idx = blockIdx.x * blockDim.x + threadIdx.x;
  if (idx < N * 4) {
    int n = idx >> 2, h = idx & 3;
    const float* xr = xs + (size_t)n * 128 + h * 32;
    const float* av = as_ + h * 32;
    const float* dv = ad_ + h * 32;
    float s1 = 0.f, s2 = 0.f;
#pragma unroll
    for (int c = 0; c < 32; ++c) {
      float v = xr[c];
      s1 += v * av[c];
      s2 += v * dv[c];
    }
    asrc[idx] = s1;
    adst[idx] = s2;
  }
}

// pass1: logits + leaky-relu, atomic segment max over dst
__global__ __launch_bounds__(256) void k_edge_pass1(
    const int* __restrict__ src, const int* __restrict__ dst,
    const float* __restrict__ ea, const float* __restrict__ asrc,
    const float* __restrict__ adst, const float* __restrict__ wd,
    float* __restrict__ alpha, unsigned* __restrict__ amax, int E, int EN) {
  int idx = blockIdx.x * blockDim.x + threadIdx.x;
  if (idx < EN * 4) {
    int e = idx >> 2, h = idx & 3;
    int s = (e < E) ? src[e] : (e - E);
    int d = (e < E) ? dst[e] : (e - E);
    float al = asrc[(size_t)s * 4 + h] + adst[(size_t)d * 4 + h] +
               ea[e] * wd[h];
    al = (al > 0.f) ? al : NEG_SLOPE * al;
    alpha[idx] = al;
    atomicMax(&amax[(size_t)d * 4 + h], fenc(al));
  }
}

// pass2: exp(alpha - max), atomic segment sum
__global__ __launch_bounds__(256) void k_edge_pass2(
    const int* __restrict__ dst, float* __restrict__ alpha,
    const unsigned* __restrict__ amax, float* __restrict__ denom, int E,
    int EN) {
  int idx = blockIdx.x * blockDim.x + threadIdx.x;
  if (idx < EN * 4) {
    int e = idx >> 2, h = idx & 3;
    int d = (e < E) ? dst[e] : (e - E);
    float mx = fdec(amax[(size_t)d * 4 + h]);
    float ex = expf(alpha[idx] - mx);
    alpha[idx] = ex;
    atomicAdd(&denom[(size_t)d * 4 + h], ex);
  }
}

// pass3: scatter xs[src]*alpha_norm into agg[dst]
// one thread per (edge, 4-channel group): 1 x b128 gather + 4 atomics
__global__ __launch_bounds__(256) void k_scatter(
    const int* __restrict__ src, const int* __restrict__ dst,
    const float* __restrict__ xs, const float* __restrict__ alpha,
    const float* __restrict__ denom, float* __restrict__ agg, int E,
    long long tot32) {
  long long idx = (long long)blockIdx.x * blockDim.x + threadIdx.x;
  if (idx < tot32) {
    int e = (int)(idx >> 5);
    int g = (int)(idx & 31);
    int h = g >> 3;
    int c = (g & 7) * 4;
    int s = (e < E) ? src[e] : (e - E);
    int d = (e < E) ? dst[e] : (e - E);
    float coef = alpha[(size_t)e * 4 + h] / (denom[(size_t)d * 4 + h] + EPS_F);
    const float4 xv = *(const float4*)(xs + (size_t)s * 128 + h * 32 + c);
    float* ap = agg + (size_t)d * 128 + h * 32 + c;
    atomicAdd(ap + 0, xv.x * coef);
    atomicAdd(ap + 1, xv.y * coef);
    atomicAdd(ap + 2, xv.z * coef);
    atomicAdd(ap + 3, xv.w * coef);
  }
}

// mean over heads + bias + relu -> out[N,32]  (float4 over channels)
__global__ __launch_bounds__(256) void k_finalize(
    const float* __restrict__ agg, const float* __restrict__ bias,
    float* __restrict__ out, int N) {
  int idx = blockIdx.x * blockDim.x + threadIdx.x;
  if (idx < N * 8) {
    int n = idx >> 3, c = (idx & 7) * 4;
    const float* ar = agg + (size_t)n * 128 + c;
    const float4 a0 = *(const float4*)(ar);
    const float4 a1 = *(const float4*)(ar + 32);
    const float4 a2 = *(const float4*)(ar + 64);
    const float4 a3 = *(const float4*)(ar + 96);
    const float4 bv = *(const float4*)(bias + c);
    float4 o;
    o.x = fmaxf(0.25f * (a0.x + a1.x + a2.x + a3.x) + bv.x, 0.f);
    o.y = fmaxf(0.25f * (a0.y + a1.y + a2.y + a3.y) + bv.y, 0.f);
    o.z = fmaxf(0.25f * (a0.z + a1.z + a2.z + a3.z) + bv.z, 0.f);
    o.w = fmaxf(0.25f * (a0.w + a1.w + a2.w + a3.w) + bv.w, 0.f);
    *(float4*)(out + (size_t)n * 32 + c) = o;
  }
}

static inline unsigned cdiv(long long a, int b) {
  return (unsigned)((a + b - 1) / b);
}

extern "C" void kernel_launch(void* const* d_in, const int* in_sizes, int n_in,
                              void* d_out, int out_size, void* d_ws,
                              size_t ws_size, hipStream_t stream) {
  const float* x   = (const float*)d_in[0];
  const int*   src = (const int*)d_in[1];
  const float* ew  = (const float*)d_in[2];
  const float* w1  = (const float*)d_in[3];
  const float* we1 = (const float*)d_in[4];
  const float* as1 = (const float*)d_in[5];
  const float* ad1 = (const float*)d_in[6];
  const float* ae1 = (const float*)d_in[7];
  const float* b1  = (const float*)d_in[8];
  const float* w2  = (const float*)d_in[9];
  const float* we2 = (const float*)d_in[10];
  const float* as2 = (const float*)d_in[11];
  const float* ad2 = (const float*)d_in[12];
  const float* ae2 = (const float*)d_in[13];
  const float* b2  = (const float*)d_in[14];
  (void)n_in; (void)out_size; (void)ws_size;

  const int N  = in_sizes[0] / 64;  // F_IN = 64
  const int E  = in_sizes[1] / 2;   // edge_index is [2,E]
  const int EN = E + N;
  const int* dst = src + E;

  // ---- workspace layout ----
  char* ws = (char*)d_ws;
  size_t off = 0;
  auto take = [&](size_t bytes) -> char* {
    char* p = ws + off;
    off = (off + bytes + 255) & ~(size_t)255;
    return p;
  };
  float*    ea    = (float*)take((size_t)EN * 4);
  float*    sums  = (float*)take((size_t)N * 4);
  float*    cnt   = (float*)take((size_t)N * 4);
  float*    wdot  = (float*)take(8 * 4);
  float*    xs    = (float*)take((size_t)N * 128 * 4);
  float*    asrc  = (float*)take((size_t)N * 4 * 4);
  float*    adst  = (float*)take((size_t)N * 4 * 4);
  unsigned* amax  = (unsigned*)take((size_t)N * 4 * 4);
  float*    denom = (float*)take((size_t)N * 4 * 4);
  float*    alpha = (float*)take((size_t)EN * 4 * 4);
  float*    agg   = (float*)take((size_t)N * 128 * 4);
  float*    h1    = (float*)take((size_t)N * 32 * 4);

  const int B = 256;
  const unsigned gemm_blocks = cdiv(((long long)N + 15) / 16, 8);
  const long long tot32 = (long long)EN * 32;

  // ---- shared precompute ----
  hipMemsetAsync(sums, 0, (size_t)N * 4, stream);
  hipMemsetAsync(cnt, 0, (size_t)N * 4, stream);
  k_edge_degree<<<cdiv(E, B), B, 0, stream>>>(dst, ew, sums, cnt, E);
  k_build_ea<<<cdiv(EN, B), B, 0, stream>>>(ew, sums, cnt, ea, E, EN);
  k_wedot<<<1, 32, 0, stream>>>(we1, ae1, we2, ae2, wdot);

  // ---- layer 1 ----
  gemm_node_wmma<64><<<gemm_blocks, B, 0, stream>>>(x, w1, xs, N);
  k_node_attn<<<cdiv((long long)N * 4, B), B, 0, stream>>>(xs, as1, ad1, asrc,
                                                           adst, N);
  hipMemsetAsync(amax, 0, (size_t)N * 16, stream);
  hipMemsetAsync(denom, 0, (size_t)N * 16, stream);
  hipMemsetAsync(agg, 0, (size_t)N * 512, stream);
  k_edge_pass1<<<cdiv((long long)EN * 4, B), B, 0, stream>>>(
      src, dst, ea, asrc, adst, wdot, alpha, amax, E, EN);
  k_edge_pass2<<<cdiv((long long)EN * 4, B), B, 0, stream>>>(dst, alpha, amax,
                                                             denom, E, EN);
  k_scatter<<<cdiv(tot32, B), B, 0, stream>>>(src, dst, xs, alpha, denom, agg,
                                              E, tot32);
  k_finalize<<<cdiv((long long)N * 8, B), B, 0, stream>>>(agg, b1, h1, N);

  // ---- layer 2 ----
  gemm_node_wmma<32><<<gemm_blocks, B, 0, stream>>>(h1, w2, xs, N);
  k_node_attn<<<cdiv((long long)N * 4, B), B, 0, stream>>>(xs, as2, ad2, asrc,
                                                           adst, N);
  hipMemsetAsync(amax, 0, (size_t)N * 16, stream);
  hipMemsetAsync(denom, 0, (size_t)N * 16, stream);
  hipMemsetAsync(agg, 0, (size_t)N * 512, stream);
  k_edge_pass1<<<cdiv((long long)EN * 4, B), B, 0, stream>>>(
      src, dst, ea, asrc, adst, wdot + 4, alpha, amax, E, EN);
  k_edge_pass2<<<cdiv((long long)EN * 4, B), B, 0, stream>>>(dst, alpha, amax,
                                                             denom, E, EN);
  k_scatter<<<cdiv(tot32, B), B, 0, stream>>>(src, dst, xs, alpha, denom, agg,
                                              E, tot32);
  k_finalize<<<cdiv((long long)N * 8, B), B, 0, stream>>>(agg, b2,
                                                          (float*)d_out, N);
}